// YOLOWithDifferentiableNMS_13030930776353
// MI455X (gfx1250) — compile-verified
//
#include <hip/hip_runtime.h>
#include <hip/hip_bf16.h>
#include <math.h>

typedef __attribute__((ext_vector_type(2))) float v2f;
typedef __attribute__((ext_vector_type(8))) float v8f;

#define CHUNK 512           // columns staged in LDS per block iteration
#define BLOCK_THREADS 256   // 8 waves (wave32)
#define WAVES_PER_BLOCK 8

// -2/0.5 temperature folded with log2(e):  exp(-2*iou) == exp2(K2*iou)
#define K2_NEG_TWO_LOG2E (-2.8853900817779268f)

// Integer min/max == float min/max for non-negative IEEE floats (bit pattern
// order). Box coords here are always >= 0, so this avoids the canonicalize
// ops LLVM inserts before v_min_num/v_max_num on memory-sourced operands.
#define IMIN(a, b) ((a) < (b) ? (a) : (b))
#define IMAX(a, b) ((a) > (b) ? (a) : (b))

// gfx1250 async global->LDS staging (existence probe-confirmed in round 2/3).
#if defined(__has_builtin)
#if __has_builtin(__builtin_amdgcn_global_load_async_to_lds_b128) && \
    __has_builtin(__builtin_amdgcn_global_load_async_to_lds_b32)  && \
    __has_builtin(__builtin_amdgcn_s_wait_asynccnt)
#define USE_ASYNC_LDS 1
#endif
#endif

#ifdef USE_ASYNC_LDS
// Exact parameter types per hipcc diagnostics:
//   b128: (AS1 int-vec4*, AS3 int-vec4*, imm, imm)
//   b32:  (AS1 int*,      AS3 int*,      imm, imm)
typedef int v4i_ __attribute__((vector_size(16)));
typedef __attribute__((address_space(1))) v4i_ gv4i;
typedef __attribute__((address_space(3))) v4i_ sv4i;
typedef __attribute__((address_space(1))) int  gi32;
typedef __attribute__((address_space(3))) int  si32;
#endif

// Phase 1: adjusted[b][i] = sum_j exp(-2*IoU(i,j)) * scores[b][j]
// One wave per 16-row tile; WMMA_F32_16x16x4_F32 accumulates row sums into
// column 0 of the f32 C/D accumulator (B = ones in column 0).
__global__ __launch_bounds__(BLOCK_THREADS)
void soft_nms_adjusted_kernel(const float* __restrict__ boxes,
                              const float* __restrict__ scores,
                              float* __restrict__ adjusted,
                              int N, int blocksPerBatch) {
    __shared__ float4 sbox[CHUNK];
    __shared__ float  ssc[CHUNK];
    __shared__ float  sarea[CHUNK];

    const int wave     = threadIdx.x >> 5;
    const int lane     = threadIdx.x & 31;
    const int laneHalf = lane >> 4;   // 0: K=0,1   1: K=2,3
    const int laneM    = lane & 15;   // row within tile (A), col within half (B/C/D)

    const int batch       = blockIdx.x / blocksPerBatch;
    const int tileInBatch = (blockIdx.x % blocksPerBatch) * WAVES_PER_BLOCK + wave;
    const int i0          = tileInBatch * 16;

    const float4* __restrict__ bb = (const float4*)boxes + (size_t)batch * N;
    const float*  __restrict__ sc = scores + (size_t)batch * N;

    // This wave-lane's fixed row box (A-matrix row M = laneM)
    const float4 rb    = bb[i0 + laneM];
    const float  areaI = (rb.z - rb.x) * (rb.w - rb.y);
    const int rx1 = __float_as_int(rb.x), ry1 = __float_as_int(rb.y);
    const int rx2 = __float_as_int(rb.z), ry2 = __float_as_int(rb.w);

    // B matrix: ones in column N=0 only -> D[:,0] = row sums of A.
    v2f bvec;
    bvec.x = (laneM == 0) ? 1.0f : 0.0f;
    bvec.y = bvec.x;

    v8f c = {};  // f32 accumulator, carries `adjusted` across the whole j loop

    for (int j0 = 0; j0 < N; j0 += CHUNK) {
        __syncthreads();   // previous chunk fully consumed
#ifdef USE_ASYNC_LDS
        for (int t = threadIdx.x; t < CHUNK; t += BLOCK_THREADS) {
            __builtin_amdgcn_global_load_async_to_lds_b128(
                (gv4i*)(bb + j0 + t), (sv4i*)&sbox[t], 0, 0);
            __builtin_amdgcn_global_load_async_to_lds_b32(
                (gi32*)(sc + j0 + t), (si32*)&ssc[t], 0, 0);
        }
        __builtin_amdgcn_s_wait_asynccnt(0);
#else
        for (int t = threadIdx.x; t < CHUNK; t += BLOCK_THREADS) {
            sbox[t] = bb[j0 + t];
            ssc[t]  = sc[j0 + t];
        }
#endif
        __syncthreads();
        // precompute column areas once per chunk (saves 3 VALU ops per pair)
        for (int t = threadIdx.x; t < CHUNK; t += BLOCK_THREADS) {
            const float4 bj = sbox[t];
            sarea[t] = (bj.z - bj.x) * (bj.w - bj.y);
        }
        __syncthreads();

        #pragma unroll 4
        for (int jj = 0; jj < CHUNK; jj += 4) {
            const int ja = jj + laneHalf * 2;   // this lane's K columns
            const float4 b0 = sbox[ja];
            const float4 b1 = sbox[ja + 1];
            const float  s0 = ssc[ja];
            const float  s1 = ssc[ja + 1];
            const float  sa0 = areaI + sarea[ja];
            const float  sa1 = areaI + sarea[ja + 1];

            // pair 0 (int min/max on non-negative float bit patterns)
            float mnx0 = __int_as_float(IMIN(rx2, __float_as_int(b0.z)));
            float mxx0 = __int_as_float(IMAX(rx1, __float_as_int(b0.x)));
            float mny0 = __int_as_float(IMIN(ry2, __float_as_int(b0.w)));
            float mxy0 = __int_as_float(IMAX(ry1, __float_as_int(b0.y)));
            float iw0  = fmaxf(0.0f, mnx0 - mxx0);
            float ih0  = fmaxf(0.0f, mny0 - mxy0);
            float in0  = iw0 * ih0;
            float un0  = sa0 - in0;
            float w0   = __builtin_amdgcn_exp2f(
                             (K2_NEG_TWO_LOG2E * in0) * __builtin_amdgcn_rcpf(un0));

            // pair 1
            float mnx1 = __int_as_float(IMIN(rx2, __float_as_int(b1.z)));
            float mxx1 = __int_as_float(IMAX(rx1, __float_as_int(b1.x)));
            float mny1 = __int_as_float(IMIN(ry2, __float_as_int(b1.w)));
            float mxy1 = __int_as_float(IMAX(ry1, __float_as_int(b1.y)));
            float iw1  = fmaxf(0.0f, mnx1 - mxx1);
            float ih1  = fmaxf(0.0f, mny1 - mxy1);
            float in1  = iw1 * ih1;
            float un1  = sa1 - in1;
            float w1   = __builtin_amdgcn_exp2f(
                             (K2_NEG_TWO_LOG2E * in1) * __builtin_amdgcn_rcpf(un1));

            // A matrix 16x4 f32: lanes 0-15 hold K=0 (v0), K=1 (v1);
            //                    lanes 16-31 hold K=2 (v0), K=3 (v1).
            v2f a;
            a.x = w0 * s0;
            a.y = w1 * s1;

            // D = A x B + C   (accumulate adjusted in column 0 of C/D)
            c = __builtin_amdgcn_wmma_f32_16x16x4_f32(
                    /*neg_a=*/false, a, /*neg_b=*/false, bvec,
                    /*c_mod=*/(short)0, c, /*reuse_a=*/false, /*reuse_b=*/false);
        }
    }

    // C/D layout: column N=0 lives in lane 0 (M=0..7 in v0..v7) and
    // lane 16 (M=8..15 in v0..v7).
    if (laneM == 0) {
        float* outp = adjusted + (size_t)batch * N + i0 + laneHalf * 8;
        #pragma unroll
        for (int r = 0; r < 8; ++r) outp[r] = c[r];
    }
}

// Phase 2+3: per batch, softmax(adjusted) then weighted sum of boxes.
__global__ __launch_bounds__(256)
void soft_nms_finalize_kernel(const float* __restrict__ adjusted,
                              const float* __restrict__ boxes,
                              float* __restrict__ out, int N) {
    const int b = blockIdx.x;
    const float*  adj = adjusted + (size_t)b * N;
    const float4* bx  = (const float4*)boxes + (size_t)b * N;

    __shared__ float red[256];

    // max reduce
    float m = -INFINITY;
    for (int t = threadIdx.x; t < N; t += blockDim.x) m = fmaxf(m, adj[t]);
    red[threadIdx.x] = m;
    __syncthreads();
    for (int s = 128; s > 0; s >>= 1) {
        if (threadIdx.x < s) red[threadIdx.x] = fmaxf(red[threadIdx.x], red[threadIdx.x + s]);
        __syncthreads();
    }
    m = red[0];
    __syncthreads();

    // exp-sum and weighted box sums
    float se = 0.f, s0 = 0.f, s1 = 0.f, s2 = 0.f, s3 = 0.f;
    for (int t = threadIdx.x; t < N; t += blockDim.x) {
        float  e = __expf(adj[t] - m);
        float4 v = bx[t];
        se += e; s0 += e * v.x; s1 += e * v.y; s2 += e * v.z; s3 += e * v.w;
    }

    float vals[5] = {se, s0, s1, s2, s3};
    float res[5];
    for (int k = 0; k < 5; ++k) {
        __syncthreads();
        red[threadIdx.x] = vals[k];
        __syncthreads();
        for (int s = 128; s > 0; s >>= 1) {
            if (threadIdx.x < s) red[threadIdx.x] += red[threadIdx.x + s];
            __syncthreads();
        }
        res[k] = red[0];
    }

    if (threadIdx.x == 0) {
        const float inv = 1.0f / res[0];
        out[b * 4 + 0] = res[1] * inv;
        out[b * 4 + 1] = res[2] * inv;
        out[b * 4 + 2] = res[3] * inv;
        out[b * 4 + 3] = res[4] * inv;
    }
}

extern "C" void kernel_launch(void* const* d_in, const int* in_sizes, int n_in,
                              void* d_out, int out_size, void* d_ws, size_t ws_size,
                              hipStream_t stream) {
    const float* boxes  = (const float*)d_in[0];  // (B, N, 4) f32
    const float* scores = (const float*)d_in[1];  // (B, N)    f32
    float* out = (float*)d_out;                   // (B, 4)    f32

    const int B = out_size / 4;          // 4
    const int N = in_sizes[1] / B;       // 8192
    float* adjusted = (float*)d_ws;      // B*N floats of scratch

    const int tilesPerBatch  = N / 16;                          // 512
    const int blocksPerBatch = tilesPerBatch / WAVES_PER_BLOCK; // 64

    dim3 grid1(B * blocksPerBatch);
    soft_nms_adjusted_kernel<<<grid1, BLOCK_THREADS, 0, stream>>>(
        boxes, scores, adjusted, N, blocksPerBatch);

    soft_nms_finalize_kernel<<<B, 256, 0, stream>>>(adjusted, boxes, out, N);
}